// mLSTMLayer_48507360641346
// MI455X (gfx1250) — compile-verified
//
#include <hip/hip_runtime.h>
#include <hip/hip_bf16.h>

#define Hdim 4096

typedef __attribute__((ext_vector_type(2))) float v2f;
typedef __attribute__((ext_vector_type(8))) float v8f;

// One 16x16x4 f32 WMMA step. A: this lane's 2 contiguous matrix elements of
// its row; B: the matching 2 vector elements (broadcast over all 16 columns).
// Lane-half K assignment is mirrored between A and B, so D[m,n] = partial
// y[m] replicated across n.
__device__ __forceinline__ v8f wmma_mv_step(float a0, float a1, float b0, float b1, v8f c) {
    v2f A; A[0] = a0; A[1] = a1;
    v2f B; B[0] = b0; B[1] = b1;
    return __builtin_amdgcn_wmma_f32_16x16x4_f32(false, A, false, B, (short)0, c,
                                                 false, false);
}

struct WB {
    const float* W[6];
    const float* b[6];
};

// ---------------------------------------------------------------------------
// K1: six projections  proj[g*H + m] = transform_g(W_g[m,:] . x + b_g[m])
// grid = 6 * (H/16) stripes; block = 256 (8 waves), K split across waves.
// ---------------------------------------------------------------------------
__global__ __launch_bounds__(256) void k_proj(WB wb, const float* __restrict__ x,
                                              float* __restrict__ proj) {
    __shared__ float lx[Hdim];
    __shared__ float red[8 * 16];
    const int tid = threadIdx.x;

    for (int i = tid; i < Hdim / 4; i += 256)
        ((float4*)lx)[i] = ((const float4*)x)[i];
    __syncthreads();

    const int stripe = blockIdx.x;
    const int g  = stripe >> 8;           // which of the 6 matrices
    const int rb = (stripe & 255) << 4;   // row base within that matrix
    const int lane = tid & 31, wv = tid >> 5;
    const int half = lane >> 4;
    const int row  = rb + (lane & 15);
    const float* Wrow = wb.W[g] + (size_t)row * Hdim;

    v8f acc = {};
    const int j0 = wv * 512;              // 512 columns per wave
    for (int j = j0; j < j0 + 512; j += 8) {
        const int col = j + half * 4;     // lanes 0-15: j..j+3, lanes 16-31: j+4..j+7
        float4 a4 = *(const float4*)(Wrow + col);
        float4 x4 = *(const float4*)(&lx[col]);
        acc = wmma_mv_step(a4.x, a4.y, x4.x, x4.y, acc);
        acc = wmma_mv_step(a4.z, a4.w, x4.z, x4.w, acc);
    }

    if (lane == 0)
        for (int r = 0; r < 8; ++r) red[wv * 16 + r] = acc[r];
    if (lane == 16)
        for (int r = 0; r < 8; ++r) red[wv * 16 + 8 + r] = acc[r];
    __syncthreads();

    if (tid < 16) {
        float s = 0.f;
        for (int w = 0; w < 8; ++w) s += red[w * 16 + tid];
        const int m = rb + tid;
        s += wb.b[g][m];
        if (g == 1) s *= 0.015625f;                 // k / sqrt(4096)
        else if (g == 5) s = 1.0f / (1.0f + expf(-s));  // sigmoid(o)
        proj[g * Hdim + m] = s;
    }
}

// ---------------------------------------------------------------------------
// K2: gates, m_t, n_t, and per-block partials of dot(n_t, q)
// ---------------------------------------------------------------------------
__global__ __launch_bounds__(256) void k_gates(const float* __restrict__ proj,
                                               const float* __restrict__ n_prev,
                                               const float* __restrict__ m_prev,
                                               float* __restrict__ out_n,
                                               float* __restrict__ out_m,
                                               float* __restrict__ ig,
                                               float* __restrict__ fg,
                                               float* __restrict__ dotp) {
    __shared__ float red[256];
    const int m = blockIdx.x * 256 + threadIdx.x;
    const float it = proj[3 * Hdim + m];
    const float ft = proj[4 * Hdim + m];
    const float mp = m_prev[m];
    const float mt  = fmaxf(ft + mp, it);
    const float igv = expf(it - mt);
    const float fgv = expf(ft + mp - mt);
    const float nt  = fgv * n_prev[m] + igv * proj[1 * Hdim + m];
    out_m[m] = mt;
    out_n[m] = nt;
    ig[m] = igv;
    fg[m] = fgv;

    red[threadIdx.x] = nt * proj[0 * Hdim + m];   // n_t * q
    __syncthreads();
    for (int s = 128; s > 0; s >>= 1) {
        if (threadIdx.x < s) red[threadIdx.x] += red[threadIdx.x + s];
        __syncthreads();
    }
    if (threadIdx.x == 0) dotp[blockIdx.x] = red[0];
}

// ---------------------------------------------------------------------------
// K3: fused  c_t = f_g*c_prev + (i_g*v)*k   and   cq = c_t @ q  (WMMA)
// grid = H/16 stripes; block = 256 (8 waves), columns split across waves.
// ---------------------------------------------------------------------------
__global__ __launch_bounds__(256) void k_cell(const float* __restrict__ c_prev,
                                              const float* __restrict__ proj,
                                              const float* __restrict__ ig,
                                              const float* __restrict__ fg,
                                              float* __restrict__ out_c,
                                              float* __restrict__ cq) {
    __shared__ float lk[Hdim];
    __shared__ float lq[Hdim];
    __shared__ float red[8 * 16];
    const int tid = threadIdx.x;

    for (int i = tid; i < Hdim / 4; i += 256) {
        ((float4*)lk)[i] = ((const float4*)(proj + 1 * Hdim))[i];
        ((float4*)lq)[i] = ((const float4*)(proj + 0 * Hdim))[i];
    }
    __syncthreads();

    const int rb = blockIdx.x << 4;
    const int lane = tid & 31, wv = tid >> 5;
    const int half = lane >> 4;
    const int row  = rb + (lane & 15);
    const float fgl = fg[row];
    const float ivl = ig[row] * proj[2 * Hdim + row];  // i_g * v
    const float* cprow = c_prev + (size_t)row * Hdim;
    float*       ctrow = out_c  + (size_t)row * Hdim;

    v8f acc = {};
    const int j0 = wv * 512;
    for (int j = j0; j < j0 + 512; j += 8) {
        const int col = j + half * 4;
        float4 cp = *(const float4*)(cprow + col);
        float4 k4 = *(const float4*)(&lk[col]);
        float4 q4 = *(const float4*)(&lq[col]);
        float4 ct;
        ct.x = fgl * cp.x + ivl * k4.x;
        ct.y = fgl * cp.y + ivl * k4.y;
        ct.z = fgl * cp.z + ivl * k4.z;
        ct.w = fgl * cp.w + ivl * k4.w;
        *(float4*)(ctrow + col) = ct;
        acc = wmma_mv_step(ct.x, ct.y, q4.x, q4.y, acc);
        acc = wmma_mv_step(ct.z, ct.w, q4.z, q4.w, acc);
    }

    if (lane == 0)
        for (int r = 0; r < 8; ++r) red[wv * 16 + r] = acc[r];
    if (lane == 16)
        for (int r = 0; r < 8; ++r) red[wv * 16 + 8 + r] = acc[r];
    __syncthreads();

    if (tid < 16) {
        float s = 0.f;
        for (int w = 0; w < 8; ++w) s += red[w * 16 + tid];
        cq[rb + tid] = s;
    }
}

// ---------------------------------------------------------------------------
// K4: h_t = o * (c_t@q) / max(|n_t.q|, 1)
// ---------------------------------------------------------------------------
__global__ __launch_bounds__(256) void k_out(const float* __restrict__ proj,
                                             const float* __restrict__ cq,
                                             const float* __restrict__ dotp,
                                             float* __restrict__ out_h) {
    const int m = blockIdx.x * 256 + threadIdx.x;
    float s = 0.f;
    for (int i = 0; i < 16; ++i) s += dotp[i];
    const float denom = fmaxf(fabsf(s), 1.0f);
    out_h[m] = proj[5 * Hdim + m] * cq[m] / denom;
}

extern "C" void kernel_launch(void* const* d_in, const int* in_sizes, int n_in,
                              void* d_out, int out_size, void* d_ws, size_t ws_size,
                              hipStream_t stream) {
    const float* x      = (const float*)d_in[0];
    // d_in[1] = h_prev (unused by the reference math)
    const float* c_prev = (const float*)d_in[2];
    const float* n_prev = (const float*)d_in[3];
    const float* m_prev = (const float*)d_in[4];

    WB wb;
    for (int g = 0; g < 6; ++g) {
        wb.W[g] = (const float*)d_in[5 + 2 * g];
        wb.b[g] = (const float*)d_in[6 + 2 * g];
    }

    // Output layout: h_t [H] | c_t [H*H] | n_t [H] | m_t [H]
    float* out   = (float*)d_out;
    float* out_h = out;
    float* out_c = out + Hdim;
    float* out_n = out + Hdim + (size_t)Hdim * Hdim;
    float* out_m = out_n + Hdim;

    // Workspace layout (floats): proj[6H] | ig[H] | fg[H] | cq[H] | dotp[16]
    float* ws   = (float*)d_ws;
    float* proj = ws;
    float* ig   = ws + 6 * Hdim;
    float* fg   = ws + 7 * Hdim;
    float* cq   = ws + 8 * Hdim;
    float* dotp = ws + 9 * Hdim;

    k_proj <<<6 * (Hdim / 16), 256, 0, stream>>>(wb, x, proj);
    k_gates<<<Hdim / 256,      256, 0, stream>>>(proj, n_prev, m_prev,
                                                 out_n, out_m, ig, fg, dotp);
    k_cell <<<Hdim / 16,       256, 0, stream>>>(c_prev, proj, ig, fg, out_c, cq);
    k_out  <<<Hdim / 256,      256, 0, stream>>>(proj, cq, dotp, out_h);
}